// GraphEncoder_21698174779630
// MI455X (gfx1250) — compile-verified
//
#include <hip/hip_runtime.h>
#include <hip/hip_bf16.h>
#include <math.h>

#define DD 128
#define HH 8
#define CC 16
#define NEG_SLOPE 0.2f
#define LN_EPS 1e-5f

typedef float v2f __attribute__((ext_vector_type(2)));
typedef float v8f __attribute__((ext_vector_type(8)));

// ---------------------------------------------------------------------------
// GEMM: C[M,128] = A[M,128] @ W[128,128] (+ bias), f32 WMMA 16x16x4.
// One wave computes a 16x32 strip: two 16x16 accumulators share one A
// fragment per K-step (halves A gather traffic, 2 WMMAs per 3 loads).
// A-frag: lanes 0-15 rows M with vgpr pair = K,K+1; lanes 16-31 K+2,K+3.
// B-frag: row K striped across lanes (N=lane&15); lanes>=16 rows K+2/K+3.
// C: vgpr v, lane<16 -> M=row0+v, lane>=16 -> M=row0+v+8, N=col0+(lane&15).
// ---------------------------------------------------------------------------
__global__ __launch_bounds__(128) void wmma_gemm_f32(
    const float* __restrict__ A, const float* __restrict__ W,
    const float* __restrict__ bias, float* __restrict__ C, int M) {
  const int wave = threadIdx.x >> 5;
  const int lane = threadIdx.x & 31;
  const int strip = blockIdx.x * 4 + wave;     // 16x32 strips
  const int strips_n = DD / 32;                // 4
  const int tm = strip / strips_n;
  const int tnp = strip - tm * strips_n;
  if (tm * 16 >= M) return;                    // wave-uniform
  const int row0 = tm * 16, col0 = tnp * 32;
  const int half = lane >> 4;
  const int l = lane & 15;

  const float* __restrict__ arow = A + (size_t)(row0 + l) * DD;
  const float* __restrict__ wcol0 = W + (size_t)(col0 + l);
  const float* __restrict__ wcol1 = W + (size_t)(col0 + 16 + l);

  v8f acc0 = {};
  v8f acc1 = {};
#pragma unroll
  for (int k = 0; k < DD; k += 4) {
    const int ka = k + 2 * half;
    v2f a;
    a.x = arow[ka];
    a.y = arow[ka + 1];
    v2f b0;
    b0.x = wcol0[(size_t)ka * DD];
    b0.y = wcol0[(size_t)(ka + 1) * DD];
    v2f b1;
    b1.x = wcol1[(size_t)ka * DD];
    b1.y = wcol1[(size_t)(ka + 1) * DD];
    acc0 = __builtin_amdgcn_wmma_f32_16x16x4_f32(
        false, a, false, b0, (short)0, acc0, false, false);
    acc1 = __builtin_amdgcn_wmma_f32_16x16x4_f32(
        false, a, false, b1, (short)0, acc1, false, false);
  }
  const float bv0 = bias ? bias[col0 + l] : 0.0f;
  const float bv1 = bias ? bias[col0 + 16 + l] : 0.0f;
#pragma unroll
  for (int v = 0; v < 8; ++v) {
    float* crow = C + (size_t)(row0 + v + 8 * half) * DD;
    crow[col0 + l] = acc0[v] + bv0;
    crow[col0 + 16 + l] = acc1[v] + bv1;
  }
}

// ---------------------------------------------------------------------------
// alpha_s / alpha_d : per (node, head) dot over 16 channels
// ---------------------------------------------------------------------------
__global__ __launch_bounds__(256) void alpha_kernel(
    const float* __restrict__ hl, const float* __restrict__ a_src,
    const float* __restrict__ a_dst, float* __restrict__ as,
    float* __restrict__ ad, int N) {
  const int gid = blockIdx.x * blockDim.x + threadIdx.x;
  if (gid >= N * HH) return;
  const int n = gid >> 3;
  const int hh = gid & 7;
  const float* __restrict__ hp = hl + (size_t)n * DD + hh * CC;
  const float* __restrict__ ws = a_src + hh * CC;
  const float* __restrict__ wd = a_dst + hh * CC;
  float ss = 0.f, sd = 0.f;
#pragma unroll
  for (int c = 0; c < CC; ++c) {
    const float xv = hp[c];
    ss += xv * ws[c];
    sd += xv * wd[c];
  }
  as[gid] = ss;
  ad[gid] = sd;
}

// ---------------------------------------------------------------------------
// per-layer scratch init: m = -inf, s = 0, acc = 0
// ---------------------------------------------------------------------------
__global__ __launch_bounds__(256) void init_layer(float* __restrict__ m,
                                                  float* __restrict__ s,
                                                  float* __restrict__ acc,
                                                  int NH, int ND) {
  const int i = blockIdx.x * blockDim.x + threadIdx.x;
  if (i < NH) {
    m[i] = -INFINITY;
    s[i] = 0.f;
  }
  if (i < ND) acc[i] = 0.f;
}

__device__ __forceinline__ float leaky(float v) {
  return v > 0.f ? v : NEG_SLOPE * v;
}

// sign-aware float atomic max (init must be -inf)
__device__ __forceinline__ void atomicMaxF32(float* addr, float val) {
  if (val >= 0.f)
    atomicMax((int*)addr, __float_as_int(val));
  else
    atomicMin((unsigned int*)addr, (unsigned int)__float_as_int(val));
}

// ---------------------------------------------------------------------------
// pass 1: segment max of leaky_relu(alpha_s[src] + alpha_d[dst]) over dst
// ---------------------------------------------------------------------------
__global__ __launch_bounds__(256) void edge_max_kernel(
    const long long* __restrict__ ei, const float* __restrict__ as,
    const float* __restrict__ ad, float* __restrict__ m, int E, int Etot) {
  const int gid = blockIdx.x * blockDim.x + threadIdx.x;
  if (gid >= Etot * HH) return;
  const int e = gid >> 3;
  const int hh = gid & 7;
  long long srcN, dstN;
  if (e < E) {
    srcN = ei[e];
    dstN = ei[(size_t)E + e];
  } else {
    srcN = dstN = e - E;  // self loop
  }
  const float v = leaky(as[srcN * HH + hh] + ad[dstN * HH + hh]);
  atomicMaxF32(&m[dstN * HH + hh], v);
}

// ---------------------------------------------------------------------------
// pass 2: segment sum of exp(e - m[dst])
// ---------------------------------------------------------------------------
__global__ __launch_bounds__(256) void edge_sum_kernel(
    const long long* __restrict__ ei, const float* __restrict__ as,
    const float* __restrict__ ad, const float* __restrict__ m,
    float* __restrict__ s, int E, int Etot) {
  const int gid = blockIdx.x * blockDim.x + threadIdx.x;
  if (gid >= Etot * HH) return;
  const int e = gid >> 3;
  const int hh = gid & 7;
  long long srcN, dstN;
  if (e < E) {
    srcN = ei[e];
    dstN = ei[(size_t)E + e];
  } else {
    srcN = dstN = e - E;
  }
  const float v = leaky(as[srcN * HH + hh] + ad[dstN * HH + hh]);
  atomicAdd(&s[dstN * HH + hh], __expf(v - m[dstN * HH + hh]));
}

// ---------------------------------------------------------------------------
// pass 3: acc[dst] += h[src] * att.  One thread per (edge, head): att is
// computed once, then 16 channels stream as 4 float4 gathers + 16 atomics.
// ---------------------------------------------------------------------------
__global__ __launch_bounds__(256) void edge_msg_kernel(
    const long long* __restrict__ ei, const float* __restrict__ as,
    const float* __restrict__ ad, const float* __restrict__ m,
    const float* __restrict__ s, const float* __restrict__ hl,
    float* __restrict__ acc, int E, int Etot) {
  const int gid = blockIdx.x * blockDim.x + threadIdx.x;
  if (gid >= Etot * HH) return;
  const int e = gid >> 3;
  const int hh = gid & 7;
  long long srcN, dstN;
  if (e < E) {
    srcN = ei[e];
    dstN = ei[(size_t)E + e];
  } else {
    srcN = dstN = e - E;
  }
  const float v = leaky(as[srcN * HH + hh] + ad[dstN * HH + hh]);
  const float att = __expf(v - m[dstN * HH + hh]) / s[dstN * HH + hh];
  const float* __restrict__ hp = hl + (size_t)srcN * DD + hh * CC;
  float* op = acc + (size_t)dstN * DD + hh * CC;
#pragma unroll
  for (int c4 = 0; c4 < CC; c4 += 4) {
    const float4 hv = *(const float4*)&hp[c4];
    atomicAdd(op + c4 + 0, hv.x * att);
    atomicAdd(op + c4 + 1, hv.y * att);
    atomicAdd(op + c4 + 2, hv.z * att);
    atomicAdd(op + c4 + 3, hv.w * att);
  }
}

// ---------------------------------------------------------------------------
// finalize: h = layernorm(acc + bias + h)  — one wave32 per row
// ---------------------------------------------------------------------------
__global__ __launch_bounds__(256) void finalize_ln_kernel(
    const float* __restrict__ acc, const float* __restrict__ bias,
    const float* __restrict__ g, const float* __restrict__ b,
    float* __restrict__ h, int N) {
  const int wave = threadIdx.x >> 5;
  const int lane = threadIdx.x & 31;
  const int row = blockIdx.x * 8 + wave;
  if (row >= N) return;
  const float* ap = acc + (size_t)row * DD;
  float* hp = h + (size_t)row * DD;
  const float4 a = *(const float4*)&ap[lane * 4];
  const float4 r = *(const float4*)&hp[lane * 4];
  const float4 bb = *(const float4*)&bias[lane * 4];
  float4 v;
  v.x = a.x + bb.x + r.x;
  v.y = a.y + bb.y + r.y;
  v.z = a.z + bb.z + r.z;
  v.w = a.w + bb.w + r.w;
  float sum = v.x + v.y + v.z + v.w;
#pragma unroll
  for (int off = 16; off >= 1; off >>= 1) sum += __shfl_xor(sum, off, 32);
  const float mu = sum * (1.f / DD);
  const float d0 = v.x - mu, d1 = v.y - mu, d2 = v.z - mu, d3 = v.w - mu;
  float vs = d0 * d0 + d1 * d1 + d2 * d2 + d3 * d3;
#pragma unroll
  for (int off = 16; off >= 1; off >>= 1) vs += __shfl_xor(vs, off, 32);
  const float rstd = rsqrtf(vs * (1.f / DD) + LN_EPS);
  const float4 gg = *(const float4*)&g[lane * 4];
  const float4 be = *(const float4*)&b[lane * 4];
  float4 o;
  o.x = d0 * rstd * gg.x + be.x;
  o.y = d1 * rstd * gg.y + be.y;
  o.z = d2 * rstd * gg.z + be.z;
  o.w = d3 * rstd * gg.w + be.w;
  *(float4*)&hp[lane * 4] = o;
}

// ---------------------------------------------------------------------------
// mean pool (stage 1: partial column sums, few atomics) + final projection
// ---------------------------------------------------------------------------
__global__ __launch_bounds__(128) void zero_pooled_kernel(float* p) {
  p[threadIdx.x] = 0.f;
}

__global__ __launch_bounds__(128) void pool_kernel(const float* __restrict__ h,
                                                   float* __restrict__ pooled,
                                                   int N) {
  const int col = threadIdx.x;
  const int n0 = blockIdx.x * 64;
  const int n1 = min(n0 + 64, N);
  float sum = 0.f;
  for (int n = n0; n < n1; ++n) sum += h[(size_t)n * DD + col];
  atomicAdd(&pooled[col], sum);
}

__global__ __launch_bounds__(128) void out_proj_kernel(
    const float* __restrict__ pooled, const float* __restrict__ w_out,
    const float* __restrict__ b_out, float* __restrict__ out, float invN) {
  const int j = threadIdx.x;
  float a = b_out[j];
#pragma unroll 8
  for (int k = 0; k < DD; ++k) a += (pooled[k] * invN) * w_out[k * DD + j];
  out[j] = a;
}

// ---------------------------------------------------------------------------
extern "C" void kernel_launch(void* const* d_in, const int* in_sizes, int n_in,
                              void* d_out, int out_size, void* d_ws,
                              size_t ws_size, hipStream_t stream) {
  const float* x = (const float*)d_in[0];
  const long long* ei = (const long long*)d_in[1];
  const float* w_in = (const float*)d_in[2];
  const float* b_in = (const float*)d_in[3];
  const float* lin_w = (const float*)d_in[4];
  const float* att_src = (const float*)d_in[5];
  const float* att_dst = (const float*)d_in[6];
  const float* gat_b = (const float*)d_in[7];
  const float* ln_g = (const float*)d_in[8];
  const float* ln_b = (const float*)d_in[9];
  const float* w_out = (const float*)d_in[10];
  const float* b_out = (const float*)d_in[11];
  float* out = (float*)d_out;

  const int N = in_sizes[0] / DD;               // 50000
  const int E = in_sizes[1] / 2;                // 800000
  const int Etot = E + N;                       // self loops appended
  const int L = in_sizes[4] / (DD * DD);        // 3

  // workspace layout (floats)
  float* ws = (float*)d_ws;
  float* h = ws;                                // N*128
  float* hl = h + (size_t)N * DD;               // N*128
  float* acc = hl + (size_t)N * DD;             // N*128
  float* as = acc + (size_t)N * DD;             // N*8
  float* ad = as + (size_t)N * HH;              // N*8
  float* mx = ad + (size_t)N * HH;              // N*8
  float* sm = mx + (size_t)N * HH;              // N*8
  float* pooled = sm + (size_t)N * HH;          // 128

  const int strips = ((N + 15) / 16) * (DD / 32);  // 16x32 strips
  const int gemm_blocks = (strips + 3) / 4;        // 4 waves / block

  // input projection: h = x @ w_in + b_in
  wmma_gemm_f32<<<gemm_blocks, 128, 0, stream>>>(x, w_in, b_in, h, N);

  for (int l = 0; l < L; ++l) {
    const float* lw = lin_w + (size_t)l * DD * DD;
    // hl = h @ lin_w
    wmma_gemm_f32<<<gemm_blocks, 128, 0, stream>>>(h, lw, nullptr, hl, N);
    alpha_kernel<<<(N * HH + 255) / 256, 256, 0, stream>>>(
        hl, att_src + l * DD, att_dst + l * DD, as, ad, N);
    init_layer<<<(N * DD + 255) / 256, 256, 0, stream>>>(mx, sm, acc, N * HH,
                                                         N * DD);
    edge_max_kernel<<<(Etot * HH + 255) / 256, 256, 0, stream>>>(ei, as, ad,
                                                                 mx, E, Etot);
    edge_sum_kernel<<<(Etot * HH + 255) / 256, 256, 0, stream>>>(
        ei, as, ad, mx, sm, E, Etot);
    edge_msg_kernel<<<(Etot * HH + 255) / 256, 256, 0, stream>>>(
        ei, as, ad, mx, sm, hl, acc, E, Etot);
    finalize_ln_kernel<<<(N + 7) / 8, 256, 0, stream>>>(
        acc, gat_b + l * DD, ln_g + l * DD, ln_b + l * DD, h, N);
  }

  zero_pooled_kernel<<<1, 128, 0, stream>>>(pooled);
  pool_kernel<<<(N + 63) / 64, 128, 0, stream>>>(h, pooled, N);
  out_proj_kernel<<<1, 128, 0, stream>>>(pooled, w_out, b_out, out,
                                         1.0f / (float)N);
}